// GraphTransformer_59090160058446
// MI455X (gfx1250) — compile-verified
//
#include <hip/hip_runtime.h>
#include <math.h>

#define NNODE 40000
#define NEDGE 640000
#define KNODE 64
#define KEDGE 16
#define DIM   128
#define NHEAD 8
#define LAYERS 2

typedef __attribute__((ext_vector_type(2))) float v2f;
typedef __attribute__((ext_vector_type(8))) float v8f;

// ---------------------------------------------------------------------------
// Generic fp32 WMMA GEMM: C[M,128] = A[M,K] @ W[K,128] (+ bias broadcast over rows)
// One 16x16 C tile per wave, 8 waves per block -> one block covers 16 rows x 128 cols.
// Uses V_WMMA_F32_16X16X4_F32 (full fp32 precision, matches reference dtype).
//   A frag (16x4): lane<16 -> K=k0,k0+1 ; lane>=16 -> K=k0+2,k0+3  (contig float2)
//   B frag (4x16): VGPR j holds row K = k0 + j + 2*(lane/16), col = lane%16
//   C/D   (16x16): VGPR r holds row m0 + r + 8*(lane/16),     col = lane%16
// ---------------------------------------------------------------------------
__global__ __launch_bounds__(256) void gemm_bias_wmma(
    const float* __restrict__ A, const float* __restrict__ W,
    const float* __restrict__ bias, float* __restrict__ C, int M, int K)
{
  const int lane = threadIdx.x & 31;
  const int wave = threadIdx.x >> 5;
  const int m0   = blockIdx.x << 4;
  const int n0   = wave << 4;
  const int half = lane >> 4;   // 0 or 1
  const int l16  = lane & 15;
  const int col  = n0 + l16;

  v8f c;
  const float bv = bias ? bias[col] : 0.0f;
#pragma unroll
  for (int r = 0; r < 8; ++r) c[r] = bv;

  const float* Arow = A + (size_t)(m0 + l16) * K + 2 * half;
  const float* Wcol = W + (size_t)(2 * half) * DIM + col;
  for (int k0 = 0; k0 < K; k0 += 4) {
    v2f a, b;
    a.x = Arow[k0 + 0];
    a.y = Arow[k0 + 1];
    b.x = Wcol[(size_t)k0 * DIM];
    b.y = Wcol[(size_t)k0 * DIM + DIM];
    c = __builtin_amdgcn_wmma_f32_16x16x4_f32(
        /*neg_a=*/false, a, /*neg_b=*/false, b,
        /*c_mod=*/(short)0, c, /*reuse_a=*/false, /*reuse_b=*/false);
  }
#pragma unroll
  for (int r = 0; r < 8; ++r)
    C[(size_t)(m0 + r + 8 * half) * DIM + col] = c[r];
}

// ---------------------------------------------------------------------------
// Attention scatter pipeline
// ---------------------------------------------------------------------------
__global__ __launch_bounds__(256) void init_attn(
    float* __restrict__ m, float* __restrict__ denom, float* __restrict__ agg)
{
  int t = blockIdx.x * blockDim.x + threadIdx.x;      // over N*128
  if (t >= NNODE * DIM) return;
  agg[t] = 0.0f;
  if (t < NNODE * NHEAD) { m[t] = -3.0e38f; denom[t] = 0.0f; }
}

// alpha[e,h] = <q[dst], k[src]+eproj> / sqrt(16);  m[dst,h] = max(...)
__global__ __launch_bounds__(256) void attn_score(
    const float* __restrict__ q, const float* __restrict__ k,
    const float* __restrict__ eproj, const int* __restrict__ src,
    const int* __restrict__ dst, float* __restrict__ alpha, float* __restrict__ m)
{
  int t = blockIdx.x * blockDim.x + threadIdx.x;      // over E*H
  if (t >= NEDGE * NHEAD) return;
  int e = t >> 3, h = t & 7;
  int s = src[e], d = dst[e];
  const float4* qp = (const float4*)(q + (size_t)d * DIM + h * 16);
  const float4* kp = (const float4*)(k + (size_t)s * DIM + h * 16);
  const float4* ep = (const float4*)(eproj + (size_t)e * DIM + h * 16);
  float acc = 0.0f;
#pragma unroll
  for (int i = 0; i < 4; ++i) {
    float4 qv = qp[i], kv = kp[i], ev = ep[i];
    acc += qv.x * (kv.x + ev.x) + qv.y * (kv.y + ev.y) +
           qv.z * (kv.z + ev.z) + qv.w * (kv.w + ev.w);
  }
  acc *= 0.25f;                                       // 1/sqrt(C), C=16
  alpha[t] = acc;
  atomicMax(&m[d * NHEAD + h], acc);                  // global_atomic_max_num_f32
}

// a = exp(alpha - m[dst]);  denom[dst,h] += a   (alpha buffer overwritten with a)
__global__ __launch_bounds__(256) void attn_exp(
    const float* __restrict__ m, const int* __restrict__ dst,
    float* __restrict__ alpha, float* __restrict__ denom)
{
  int t = blockIdx.x * blockDim.x + threadIdx.x;      // over E*H
  if (t >= NEDGE * NHEAD) return;
  int e = t >> 3, h = t & 7;
  int d = dst[e];
  float a = __expf(alpha[t] - m[d * NHEAD + h]);
  alpha[t] = a;
  atomicAdd(&denom[d * NHEAD + h], a);
}

// agg[dst, c] += (v[src,c] + eproj[e,c]) * a[e,h]/(denom[dst,h]+1e-16)
__global__ __launch_bounds__(256) void attn_agg(
    const float* __restrict__ v, const float* __restrict__ eproj,
    const float* __restrict__ alpha, const float* __restrict__ denom,
    const int* __restrict__ src, const int* __restrict__ dst,
    float* __restrict__ agg)
{
  int t = blockIdx.x * blockDim.x + threadIdx.x;      // over E*128 (81.92M < 2^31)
  if (t >= NEDGE * DIM) return;
  int e = t >> 7, c = t & 127, h = c >> 4;
  int s = src[e], d = dst[e];
  float w = alpha[e * NHEAD + h] / (denom[d * NHEAD + h] + 1e-16f);
  float msg = (v[(size_t)s * DIM + c] + eproj[(size_t)t]) * w;
  atomicAdd(&agg[(size_t)d * DIM + c], msg);
}

// h += relu(agg + skip)
__global__ __launch_bounds__(256) void residual_relu(
    float* __restrict__ h, const float* __restrict__ agg,
    const float* __restrict__ skip)
{
  int t = blockIdx.x * blockDim.x + threadIdx.x;      // over N*128
  if (t >= NNODE * DIM) return;
  float hn = agg[t] + skip[t];
  h[t] += hn > 0.0f ? hn : 0.0f;
}

// ---------------------------------------------------------------------------
// LayerNorm (one wave32 per node) fused with mean pool (atomicAdd * 1/N)
// ---------------------------------------------------------------------------
__global__ __launch_bounds__(128) void init_pooled(float* __restrict__ pooled)
{
  pooled[threadIdx.x] = 0.0f;
}

__global__ __launch_bounds__(256) void ln_pool(
    const float* __restrict__ h, const float* __restrict__ gamma,
    const float* __restrict__ beta, float* __restrict__ pooled)
{
  int wave = threadIdx.x >> 5, lane = threadIdx.x & 31;
  int node = blockIdx.x * 8 + wave;
  if (node >= NNODE) return;
  float4 v = ((const float4*)(h + (size_t)node * DIM))[lane];
  float s = v.x + v.y + v.z + v.w;
#pragma unroll
  for (int off = 16; off; off >>= 1) s += __shfl_xor(s, off, 32);
  float mu = s * (1.0f / 128.0f);
  float dx = v.x - mu, dy = v.y - mu, dz = v.z - mu, dw = v.w - mu;
  float vs = dx * dx + dy * dy + dz * dz + dw * dw;
#pragma unroll
  for (int off = 16; off; off >>= 1) vs += __shfl_xor(vs, off, 32);
  float rstd = rsqrtf(vs * (1.0f / 128.0f) + 1e-5f);
  float4 g = ((const float4*)gamma)[lane];
  float4 b = ((const float4*)beta)[lane];
  const float inv = 1.0f / (float)NNODE;
  int c = lane * 4;
  atomicAdd(&pooled[c + 0], (dx * rstd * g.x + b.x) * inv);
  atomicAdd(&pooled[c + 1], (dy * rstd * g.y + b.y) * inv);
  atomicAdd(&pooled[c + 2], (dz * rstd * g.z + b.z) * inv);
  atomicAdd(&pooled[c + 3], (dw * rstd * g.w + b.w) * inv);
}

// out[j] = sum_i pooled[i] * Wout[i,j] + bout[j]   (1x128 @ 128x128)
__global__ __launch_bounds__(128) void out_proj(
    const float* __restrict__ pooled, const float* __restrict__ Wout,
    const float* __restrict__ bout, float* __restrict__ out)
{
  int j = threadIdx.x;
  float acc = bout[j];
  for (int i = 0; i < DIM; ++i) acc += pooled[i] * Wout[i * DIM + j];
  out[j] = acc;
}

// ---------------------------------------------------------------------------
extern "C" void kernel_launch(void* const* d_in, const int* in_sizes, int n_in,
                              void* d_out, int out_size, void* d_ws, size_t ws_size,
                              hipStream_t stream)
{
  const float* x         = (const float*)d_in[0];
  const float* edge_attr = (const float*)d_in[1];
  const int*   edge_idx  = (const int*)d_in[2];
  const float* W_node    = (const float*)d_in[3];
  const float* b_node    = (const float*)d_in[4];
  const float* W_ee      = (const float*)d_in[5];
  const float* b_ee      = (const float*)d_in[6];
  const float* Wq        = (const float*)d_in[7];
  const float* bq        = (const float*)d_in[8];
  const float* Wk        = (const float*)d_in[9];
  const float* bk        = (const float*)d_in[10];
  const float* Wv        = (const float*)d_in[11];
  const float* bv        = (const float*)d_in[12];
  const float* We        = (const float*)d_in[13];
  const float* Wskip     = (const float*)d_in[14];
  const float* bskip     = (const float*)d_in[15];
  const float* gamma     = (const float*)d_in[16];
  const float* beta      = (const float*)d_in[17];
  const float* Wout      = (const float*)d_in[18];
  const float* bout      = (const float*)d_in[19];

  const int* src = edge_idx;
  const int* dst = edge_idx + NEDGE;

  // ---- workspace carve-up (all offsets multiples of 128 floats -> 512B align)
  float* ws = (float*)d_ws;
  size_t off = 0;
  float* h      = ws + off; off += (size_t)NNODE * DIM;
  float* q      = ws + off; off += (size_t)NNODE * DIM;
  float* k      = ws + off; off += (size_t)NNODE * DIM;
  float* v      = ws + off; off += (size_t)NNODE * DIM;
  float* skip   = ws + off; off += (size_t)NNODE * DIM;
  float* agg    = ws + off; off += (size_t)NNODE * DIM;
  float* m      = ws + off; off += (size_t)NNODE * NHEAD;
  float* denom  = ws + off; off += (size_t)NNODE * NHEAD;
  float* e_emb  = ws + off; off += (size_t)NEDGE * DIM;
  float* e_proj = ws + off; off += (size_t)NEDGE * DIM;
  float* alpha  = ws + off; off += (size_t)NEDGE * NHEAD;
  float* pooled = ws + off; off += 128;

  const int TB = 256;
  const int gNodeTiles = NNODE / 16;                 // 2500
  const int gEdgeTiles = NEDGE / 16;                 // 40000
  const int gNElem = (NNODE * DIM + TB - 1) / TB;    // 20000
  const int gEH    = (NEDGE * NHEAD + TB - 1) / TB;  // 20000
  const int gED    = (NEDGE * DIM + TB - 1) / TB;    // 320000

  // Embeddings
  gemm_bias_wmma<<<gNodeTiles, TB, 0, stream>>>(x, W_node, b_node, h, NNODE, KNODE);
  gemm_bias_wmma<<<gEdgeTiles, TB, 0, stream>>>(edge_attr, W_ee, b_ee, e_emb, NEDGE, KEDGE);

  for (int l = 0; l < LAYERS; ++l) {
    const float* Wq_l = Wq + (size_t)l * DIM * DIM;  const float* bq_l = bq + (size_t)l * DIM;
    const float* Wk_l = Wk + (size_t)l * DIM * DIM;  const float* bk_l = bk + (size_t)l * DIM;
    const float* Wv_l = Wv + (size_t)l * DIM * DIM;  const float* bv_l = bv + (size_t)l * DIM;
    const float* We_l = We + (size_t)l * DIM * DIM;
    const float* Ws_l = Wskip + (size_t)l * DIM * DIM;
    const float* bs_l = bskip + (size_t)l * DIM;

    gemm_bias_wmma<<<gNodeTiles, TB, 0, stream>>>(h, Wq_l, bq_l, q, NNODE, DIM);
    gemm_bias_wmma<<<gNodeTiles, TB, 0, stream>>>(h, Wk_l, bk_l, k, NNODE, DIM);
    gemm_bias_wmma<<<gNodeTiles, TB, 0, stream>>>(h, Wv_l, bv_l, v, NNODE, DIM);
    gemm_bias_wmma<<<gNodeTiles, TB, 0, stream>>>(h, Ws_l, bs_l, skip, NNODE, DIM);
    gemm_bias_wmma<<<gEdgeTiles, TB, 0, stream>>>(e_emb, We_l, nullptr, e_proj, NEDGE, DIM);

    init_attn<<<gNElem, TB, 0, stream>>>(m, denom, agg);
    attn_score<<<gEH, TB, 0, stream>>>(q, k, e_proj, src, dst, alpha, m);
    attn_exp<<<gEH, TB, 0, stream>>>(m, dst, alpha, denom);
    attn_agg<<<gED, TB, 0, stream>>>(v, e_proj, alpha, denom, src, dst, agg);
    residual_relu<<<gNElem, TB, 0, stream>>>(h, agg, skip);
  }

  init_pooled<<<1, 128, 0, stream>>>(pooled);
  ln_pool<<<NNODE / 8, TB, 0, stream>>>(h, gamma, beta, pooled);
  out_proj<<<1, 128, 0, stream>>>(pooled, Wout, bout, (float*)d_out);
}